// SpatialEncodingPlus_27401891349079
// MI455X (gfx1250) — compile-verified
//
#include <hip/hip_runtime.h>

// ---------------------------------------------------------------------------
// SpatialEncodingPlus on MI455X (gfx1250, wave32, WMMA)
//
// spatial = sum_i b[i] * (A_i - A_{i-1}),  A_{i+1} = min(A_i @ A_i, 1)
// 4 fp32 GEMMs of 4096^3 (~550 GFLOP) -> compute-bound on V_WMMA_F32_16X16X4_F32.
// All matrices (64 MB) fit in the 192 MB L2; HBM traffic is negligible.
//
// v2: B tile staged TRANSPOSED in LDS so every fragment (A and B) is a single
//     8B-aligned ds_load_b64 into an adjacent VGPR pair (no v_mov repacking),
//     and wave tile raised to 64x64 (4x4 WMMA tiles) -> 16 wmma per 8 ds loads.
// ---------------------------------------------------------------------------

typedef float v2f __attribute__((ext_vector_type(2)));
typedef float v8f __attribute__((ext_vector_type(8)));

#define NN   4096          // num_nodes (compile-time, matches reference)
#define BM   256           // block tile M
#define BN   128           // block tile N
#define BK   32            // K slab staged through LDS
#define LDA  36            // As row stride (even -> 8B-aligned b64, conflict-free)
#define LDB  34            // BsT row stride (even -> 8B-aligned b64)

// ---------------- elementwise helpers ----------------

__global__ __launch_bounds__(256) void zero_f4(float* __restrict__ p) {
    size_t i = (size_t)blockIdx.x * blockDim.x + threadIdx.x;   // one float4 each
    ((float4*)p)[i] = make_float4(0.f, 0.f, 0.f, 0.f);
}

__global__ __launch_bounds__(256) void scatter_edges(const int* __restrict__ ei,
                                                     float* __restrict__ A,
                                                     int num_edges) {
    int e = blockIdx.x * blockDim.x + threadIdx.x;
    if (e < num_edges) {
        int r = ei[e];
        int c = ei[num_edges + e];
        atomicAdd(A + (size_t)r * NN + c, 1.0f);
    }
}

__global__ __launch_bounds__(256) void add_identity(float* __restrict__ A) {
    int i = blockIdx.x * blockDim.x + threadIdx.x;
    if (i < NN) A[(size_t)i * NN + i] += 1.0f;
}

// spatial = A * b[0]   (first band: adj_last == 0)
__global__ __launch_bounds__(256) void spatial_init(const float* __restrict__ A,
                                                    float* __restrict__ Sp,
                                                    const float* __restrict__ bvec) {
    size_t i = (size_t)blockIdx.x * blockDim.x + threadIdx.x;
    float b0 = bvec[0];
    float4 v = ((const float4*)A)[i];
    ((float4*)Sp)[i] = make_float4(v.x * b0, v.y * b0, v.z * b0, v.w * b0);
}

// ---------------- fused GEMM: Anext = min(A@A,1); Sp += (Anext - A)*b[bi] ----

__global__ __launch_bounds__(256)
void gemm_square_fuse(const float* __restrict__ A,
                      float* __restrict__ Anext,
                      float* __restrict__ Sp,
                      const float* __restrict__ bvec,
                      int bi) {
    __shared__ float As [BM * LDA];   // 256 x 32 row-slab of A     (~36 KB)
    __shared__ float BsT[BN * LDB];   // 128 x 32 col-slab, TRANSPOSED (~17 KB)

    const int tid    = threadIdx.x;
    const int lane   = tid & 31;
    const int lane16 = lane & 15;
    const int khalf  = (lane >> 4) << 1;   // 0 for lanes 0-15, 2 for lanes 16-31
    const int w      = tid >> 5;           // wave id 0..7
    const int wm     = w & 3;              // 4 waves along M -> 64 rows each
    const int wn     = w >> 2;             // 2 waves along N -> 64 cols each
    const int row0   = blockIdx.y * BM;
    const int col0   = blockIdx.x * BN;

    v8f acc[4][4];                          // 64x64 wave tile = 4x4 WMMA tiles
    #pragma unroll
    for (int tm = 0; tm < 4; ++tm)
        #pragma unroll
        for (int tn = 0; tn < 4; ++tn)
            acc[tm][tn] = (v8f){0.f, 0.f, 0.f, 0.f, 0.f, 0.f, 0.f, 0.f};

    for (int k0 = 0; k0 < NN; k0 += BK) {
        // Stage A row-slab  A[row0 .. row0+255][k0 .. k0+31]  (8 float4 / thread)
        #pragma unroll
        for (int i = 0; i < 8; ++i) {
            int lin = tid + i * 256;
            int r   = lin >> 3;          // 0..255
            int c4  = lin & 7;           // float4 column 0..7
            float4 v = *(const float4*)(A + (size_t)(row0 + r) * NN + k0 + c4 * 4);
            *(float4*)(&As[r * LDA + c4 * 4]) = v;
        }
        // Stage B col-slab  A[k0 .. k0+31][col0 .. col0+127], transposed into BsT
        #pragma unroll
        for (int i = 0; i < 4; ++i) {
            int lin = tid + i * 256;
            int r   = lin >> 5;          // k within slab, 0..31
            int c4  = lin & 31;          // float4 column 0..31
            float4 v = *(const float4*)(A + (size_t)(k0 + r) * NN + col0 + c4 * 4);
            BsT[(c4 * 4 + 0) * LDB + r] = v.x;
            BsT[(c4 * 4 + 1) * LDB + r] = v.y;
            BsT[(c4 * 4 + 2) * LDB + r] = v.z;
            BsT[(c4 * 4 + 3) * LDB + r] = v.w;
        }
        __syncthreads();

        #pragma unroll
        for (int kk = 0; kk < BK; kk += 4) {
            // A fragment (16x4): lanes 0-15 -> K = kk+{0,1}; lanes 16-31 -> kk+{2,3}
            v2f afrag[4];
            #pragma unroll
            for (int tm = 0; tm < 4; ++tm) {
                int m = wm * 64 + tm * 16 + lane16;
                afrag[tm] = *(const v2f*)(&As[m * LDA + kk + khalf]);   // ds_load_b64
            }
            // B fragment (4x16) from transposed tile: adjacent K pair per lane
            v2f bfrag[4];
            #pragma unroll
            for (int tn = 0; tn < 4; ++tn) {
                int n = wn * 64 + tn * 16 + lane16;
                bfrag[tn] = *(const v2f*)(&BsT[n * LDB + kk + khalf]);  // ds_load_b64
            }
            #pragma unroll
            for (int tm = 0; tm < 4; ++tm)
                #pragma unroll
                for (int tn = 0; tn < 4; ++tn)
                    acc[tm][tn] = __builtin_amdgcn_wmma_f32_16x16x4_f32(
                        /*neg_a=*/false, afrag[tm],
                        /*neg_b=*/false, bfrag[tn],
                        /*c_mod=*/(short)0, acc[tm][tn],
                        /*reuse_a=*/false, /*reuse_b=*/false);
        }
        __syncthreads();
    }

    // Epilogue: clamp, emit Anext, accumulate hop band into spatial.
    const float bs = bvec[bi];
    #pragma unroll
    for (int tm = 0; tm < 4; ++tm) {
        #pragma unroll
        for (int tn = 0; tn < 4; ++tn) {
            // C/D layout: VGPR r -> M = r (lanes 0-15) or r+8 (lanes 16-31), N = lane16
            int rbase = row0 + wm * 64 + tm * 16 + (lane >> 4) * 8;
            int col   = col0 + wn * 64 + tn * 16 + lane16;
            #pragma unroll
            for (int r = 0; r < 8; ++r) {
                size_t idx = (size_t)(rbase + r) * NN + col;
                float c    = fminf(acc[tm][tn][r], 1.0f);
                float aold = A[idx];
                Anext[idx] = c;
                Sp[idx]   += (c - aold) * bs;
            }
        }
    }
}

// ---------------- host-side orchestration ----------------

extern "C" void kernel_launch(void* const* d_in, const int* in_sizes, int n_in,
                              void* d_out, int out_size, void* d_ws, size_t ws_size,
                              hipStream_t stream) {
    const int*   ei   = (const int*)d_in[0];        // edge_index [2, E]
    const float* bvec = (const float*)d_in[2];      // b [nsteps]
    const int    num_edges = in_sizes[0] / 2;
    const int    nsteps    = in_sizes[2];

    float* A  = (float*)d_ws;                       // 64 MB
    float* A2 = (float*)d_ws + (size_t)NN * NN;     // 64 MB
    float* Sp = (float*)d_out;

    const size_t nelem   = (size_t)NN * NN;
    const int    ew_grid = (int)(nelem / 4 / 256);

    // Build A = scatter(edges) + I
    zero_f4<<<ew_grid, 256, 0, stream>>>(A);
    scatter_edges<<<(num_edges + 255) / 256, 256, 0, stream>>>(ei, A, num_edges);
    add_identity<<<(NN + 255) / 256, 256, 0, stream>>>(A);

    // spatial = A * b[0]
    spatial_init<<<ew_grid, 256, 0, stream>>>(A, Sp, bvec);

    // 4 chained squarings, hop-band update fused into the GEMM epilogue.
    float* cur = A;
    float* nxt = A2;
    dim3 ggrid(NN / BN, NN / BM);
    for (int i = 1; i < nsteps; ++i) {
        gemm_square_fuse<<<ggrid, 256, 0, stream>>>(cur, nxt, Sp, bvec, i);
        float* t = cur; cur = nxt; nxt = t;
    }
}